// SparseGraphAttentionNetwork_78288663871797
// MI455X (gfx1250) — compile-verified
//
#include <hip/hip_runtime.h>
#include <math.h>
#include <stdint.h>

// ---------------- problem constants ----------------
#define ND    256          // input feature dim
#define HK    64           // per-head hidden dim
#define NH    8            // heads
#define HC    512          // NH*HK
#define GAT_ALPHA 0.2f

typedef float v2f __attribute__((ext_vector_type(2)));
typedef float v8f __attribute__((ext_vector_type(8)));
typedef int   v4i __attribute__((ext_vector_type(4)));

#define AS1 __attribute__((address_space(1)))
#define AS3 __attribute__((address_space(3)))

#if defined(__has_builtin)
#if __has_builtin(__builtin_amdgcn_global_load_async_to_lds_b128) && \
    __has_builtin(__builtin_amdgcn_s_wait_asynccnt)
#define HAVE_ASYNC_LDS 1
#endif
#endif
#ifndef HAVE_ASYNC_LDS
#define HAVE_ASYNC_LDS 0
#endif

#if HAVE_ASYNC_LDS
// gfx1250 async global->LDS DMA (ASYNCcnt-tracked). Builtin wants
// (v4i AS1* src, v4i AS3* dst, imm offset, imm cpol); build the address-space
// pointers by value-reinterpreting through uintptr_t (flat-global addresses are
// identity mapped; LDS offset is the low 32 bits of the generic pointer).
__device__ __forceinline__ void async_copy_b128(const void* gsrc, void* ldst) {
  AS1 v4i* g = (AS1 v4i*)(uintptr_t)gsrc;
  AS3 v4i* l = (AS3 v4i*)(uint32_t)(uintptr_t)ldst;
  __builtin_amdgcn_global_load_async_to_lds_b128(g, l, 0, 0);
}
#endif

__device__ __forceinline__ float lrelu(float v) { return v > 0.0f ? v : GAT_ALPHA * v; }

// ---------------- generic fp32 WMMA GEMM: C[M,Nc] = A[M,K] @ B[K,Nc] (+bias) ----
// block = 256 threads (8 waves). Workgroup tile 128(M) x 64(N); wave w owns rows
// [w*16, w*16+16) x 64 cols as 4 accumulator fragments. K staged 32-deep in LDS
// via gfx1250 async global->LDS DMA when available.
#define TM 128
#define TN 64
#define TK 32
#define ASTR 36   // padded LDS row stride for A: 144B rows (16B aligned)
#define BSTR 68   // padded LDS row stride for B: 272B rows (16B aligned)

__launch_bounds__(256)
__global__ void gemm_f32_wmma(const float* __restrict__ A, const float* __restrict__ B,
                              const float* __restrict__ bias, float* __restrict__ C,
                              int M, int K, int Nc) {
  __shared__ float As[TM * ASTR];   // 18.0 KB
  __shared__ float Bs[TK * BSTR];   //  8.5 KB
  const int tid  = threadIdx.x;
  const int lane = tid & 31;
  const int wave = tid >> 5;         // 0..7
  const int half = lane >> 4;        // 0/1 (lane half-select in WMMA layouts)
  const int lr   = lane & 15;
  const int m0 = blockIdx.x * TM;
  const int n0 = blockIdx.y * TN;

  v8f acc[4] = {};

  for (int k0 = 0; k0 < K; k0 += TK) {
#if HAVE_ASYNC_LDS
    // ---- CDNA5 async DMA staging (no VGPR round trip) ----
    #pragma unroll
    for (int i = 0; i < (TM * TK / 4) / 256; ++i) {
      int li = tid + i * 256;
      int r  = li >> 3;              // 8 float4 per row
      int c4 = (li & 7) << 2;
      int gr = m0 + r;
      if (gr < M) {
        async_copy_b128(A + (size_t)gr * K + k0 + c4, &As[r * ASTR + c4]);
      } else {
        *(float4*)(&As[r * ASTR + c4]) = make_float4(0.f, 0.f, 0.f, 0.f);
      }
    }
    #pragma unroll
    for (int i = 0; i < (TK * TN / 4) / 256; ++i) {
      int li = tid + i * 256;
      int r  = li >> 4;              // 16 float4 per row
      int c4 = (li & 15) << 2;
      async_copy_b128(B + (size_t)(k0 + r) * Nc + n0 + c4, &Bs[r * BSTR + c4]);
    }
    __builtin_amdgcn_s_wait_asynccnt(0);
    __syncthreads();
#else
    // ---- fallback: load to VGPRs then ds_store ----
    #pragma unroll
    for (int i = 0; i < (TM * TK / 4) / 256; ++i) {
      int li = tid + i * 256;
      int r  = li >> 3;
      int c4 = (li & 7) << 2;
      int gr = m0 + r;
      float4 v = make_float4(0.f, 0.f, 0.f, 0.f);
      if (gr < M) v = *(const float4*)(A + (size_t)gr * K + k0 + c4);
      *(float4*)(&As[r * ASTR + c4]) = v;
    }
    #pragma unroll
    for (int i = 0; i < (TK * TN / 4) / 256; ++i) {
      int li = tid + i * 256;
      int r  = li >> 4;
      int c4 = (li & 15) << 2;
      float4 v = *(const float4*)(B + (size_t)(k0 + r) * Nc + n0 + c4);
      *(float4*)(&Bs[r * BSTR + c4]) = v;
    }
    __syncthreads();
#endif

    #pragma unroll
    for (int kk = 0; kk < TK; kk += 4) {
      // A frag 16x4: lanes0-15 rows, VGPR v -> K = kk + 2*half + v
      v2f a;
      a.x = As[(wave * 16 + lr) * ASTR + kk + 2 * half + 0];
      a.y = As[(wave * 16 + lr) * ASTR + kk + 2 * half + 1];
      #pragma unroll
      for (int t = 0; t < 4; ++t) {
        // B frag 4x16: VGPR v -> row K = kk + 2*half + v, col = lane%16
        v2f b;
        b.x = Bs[(kk + 2 * half + 0) * BSTR + t * 16 + lr];
        b.y = Bs[(kk + 2 * half + 1) * BSTR + t * 16 + lr];
        acc[t] = __builtin_amdgcn_wmma_f32_16x16x4_f32(
            false, a, false, b, (short)0, acc[t], false, false);
      }
    }
    __syncthreads();
  }

  // epilogue: C frag VGPR r -> row = r + 8*half, col = lane%16
  const int rb = m0 + wave * 16 + 8 * half;
  #pragma unroll
  for (int t = 0; t < 4; ++t) {
    int col = n0 + t * 16 + lr;
    float bv = bias ? bias[col] : 0.0f;
    #pragma unroll
    for (int r = 0; r < 8; ++r) {
      int row = rb + r;
      if (row < M) C[(size_t)row * Nc + col] = acc[t][r] + bv;
    }
  }
}

// ---------------- pack heads_W [H,D,K] -> row-major [D, H*K] ----------------
__global__ void pack_headsW(const float* __restrict__ hw, float* __restrict__ wpk) {
  int i = blockIdx.x * blockDim.x + threadIdx.x;
  if (i >= ND * HC) return;
  int d = i >> 9;
  int c = i & (HC - 1);
  int h = c >> 6;
  int k = c & 63;
  wpk[i] = hw[(size_t)h * (ND * HK) + (size_t)d * HK + k];
}

// ---------------- per-node attention projections ----------------
__global__ void proj_kernel(const float* __restrict__ hmat, const float* __restrict__ avec,
                            float* __restrict__ ssrc, float* __restrict__ sdst,
                            int Nn, int HH) {
  int idx = blockIdx.x * blockDim.x + threadIdx.x;
  if (idx >= Nn * HH) return;
  int n = idx / HH, h = idx - n * HH;
  const float4* hp = (const float4*)(hmat + (size_t)n * (HH * 64) + h * 64);
  const float4* as = (const float4*)(avec + h * 128);
  const float4* ad = (const float4*)(avec + h * 128 + 64);
  float s1 = 0.f, s2 = 0.f;
  #pragma unroll
  for (int i = 0; i < 16; ++i) {
    float4 v = hp[i], a1 = as[i], a2 = ad[i];
    s1 += v.x * a1.x + v.y * a1.y + v.z * a1.z + v.w * a1.w;
    s2 += v.x * a2.x + v.y * a2.y + v.z * a2.z + v.w * a2.w;
  }
  ssrc[(size_t)h * Nn + n] = s1;
  sdst[(size_t)h * Nn + n] = s2;
}

// ---------------- counting sort of edges by destination ----------------
__global__ void edge_count_kernel(const int* __restrict__ dst, int* __restrict__ cnt, int E) {
  int e = blockIdx.x * blockDim.x + threadIdx.x;
  if (e < E) atomicAdd(&cnt[dst[e]], 1);
}

// single-block exclusive scan of cnt[0..Nn) -> offs[0..Nn], offs[Nn] = E
__launch_bounds__(1024)
__global__ void scan_kernel(const int* __restrict__ cnt, int* __restrict__ offs, int Nn) {
  __shared__ int sdata[1024];
  const int tid = threadIdx.x;
  int carry = 0;
  for (int base = 0; base < Nn; base += 1024) {
    int i = base + tid;
    int v = (i < Nn) ? cnt[i] : 0;
    sdata[tid] = v;
    __syncthreads();
    for (int o = 1; o < 1024; o <<= 1) {
      int t = (tid >= o) ? sdata[tid - o] : 0;
      __syncthreads();
      sdata[tid] += t;
      __syncthreads();
    }
    int incl = sdata[tid];
    if (i < Nn) offs[i] = carry + incl - v;
    int total = sdata[1023];
    __syncthreads();
    carry += total;
  }
  if (tid == 0) offs[Nn] = carry;
}

__global__ void edge_scatter_kernel(const int* __restrict__ src, const int* __restrict__ dst,
                                    const int* __restrict__ offs, int* __restrict__ cur,
                                    int* __restrict__ sortedSrc, int E) {
  int e = blockIdx.x * blockDim.x + threadIdx.x;
  if (e >= E) return;
  int d = dst[e];
  int p = offs[d] + atomicAdd(&cur[d], 1);
  sortedSrc[p] = src[e];
}

// ---------------- segment softmax stats (no atomics): thread per (d,h) -------
__global__ void seg_softmax_kernel(const int* __restrict__ offs, const int* __restrict__ sortedSrc,
                                   const float* __restrict__ ssrc, const float* __restrict__ sdst,
                                   float* __restrict__ mArr, float* __restrict__ denArr,
                                   int Nn, int HH) {
  int idx = blockIdx.x * blockDim.x + threadIdx.x;
  if (idx >= Nn * HH) return;
  int d = idx % Nn, h = idx / Nn;               // consecutive threads -> consecutive d
  int beg = offs[d], end = offs[d + 1];
  float sd = sdst[(size_t)h * Nn + d];
  float mx = -INFINITY;
  for (int p = beg; p < end; ++p)
    mx = fmaxf(mx, lrelu(ssrc[(size_t)h * Nn + sortedSrc[p]] + sd));
  float den = 0.f;
  for (int p = beg; p < end; ++p)
    den += __expf(lrelu(ssrc[(size_t)h * Nn + sortedSrc[p]] + sd) - mx);
  mArr[(size_t)h * Nn + d] = mx;
  denArr[(size_t)h * Nn + d] = den;
}

// ---------------- atomic-free segment aggregation: block per destination -----
// 256 threads cover C = HH*64 channels (acc in regs); edges consumed in chunks
// of 32 with attention coefficients staged in LDS. Optionally fuses ELU.
__launch_bounds__(256)
__global__ void seg_agg_kernel(const int* __restrict__ offs, const int* __restrict__ sortedSrc,
                               const float* __restrict__ ssrc, const float* __restrict__ sdst,
                               const float* __restrict__ mArr, const float* __restrict__ denArr,
                               const float* __restrict__ hmat, float* __restrict__ outagg,
                               int Nn, int HH, int do_elu) {
  const int C = HH * 64;
  const int d = blockIdx.x;
  const int tid = threadIdx.x;
  __shared__ float attnS[32 * 8];
  __shared__ int   srcS[32];
  const int beg = offs[d], end = offs[d + 1];
  float acc0 = 0.f, acc1 = 0.f;

  for (int base = beg; base < end; base += 32) {
    int m = min(32, end - base);
    if (tid < 32) srcS[tid] = (tid < m) ? sortedSrc[base + tid] : 0;
    __syncthreads();
    {
      int j = tid >> 3;                 // edge-in-chunk 0..31
      int h = tid & 7;                  // head
      if (h < HH && j < m) {
        int s = srcS[j];
        float v = lrelu(ssrc[(size_t)h * Nn + s] + sdst[(size_t)h * Nn + d]);
        attnS[j * 8 + h] = __expf(v - mArr[(size_t)h * Nn + d]) /
                           (denArr[(size_t)h * Nn + d] + 1e-16f);
      }
    }
    __syncthreads();
    for (int j = 0; j < m; ++j) {
      int s = srcS[j];
      if (tid < C) acc0 += attnS[j * 8 + (tid >> 6)] * hmat[(size_t)s * C + tid];
      if (C > 256) acc1 += attnS[j * 8 + ((tid + 256) >> 6)] * hmat[(size_t)s * C + tid + 256];
    }
    __syncthreads();
  }

  if (tid < C) {
    float v = acc0;
    if (do_elu) v = v > 0.f ? v : (__expf(v) - 1.f);
    outagg[(size_t)d * C + tid] = v;
  }
  if (C > 256) {
    float v = acc1;
    if (do_elu) v = v > 0.f ? v : (__expf(v) - 1.f);
    outagg[(size_t)d * C + tid + 256] = v;
  }
}

// wave per node: ELU then row softmax over 64 outputs (2 per lane)
__global__ void elu_softmax_kernel(const float* __restrict__ agg, float* __restrict__ out, int Nn) {
  int lane = threadIdx.x & 31;
  int wid = (blockIdx.x * blockDim.x + threadIdx.x) >> 5;
  if (wid >= Nn) return;
  const float* row = agg + (size_t)wid * 64;
  float v0 = row[lane], v1 = row[lane + 32];
  v0 = v0 > 0.f ? v0 : (__expf(v0) - 1.f);
  v1 = v1 > 0.f ? v1 : (__expf(v1) - 1.f);
  float mx = fmaxf(v0, v1);
  for (int o = 16; o; o >>= 1) mx = fmaxf(mx, __shfl_xor(mx, o, 32));
  float e0 = __expf(v0 - mx), e1 = __expf(v1 - mx);
  float sm = e0 + e1;
  for (int o = 16; o; o >>= 1) sm += __shfl_xor(sm, o, 32);
  float inv = 1.0f / sm;
  out[(size_t)wid * 64 + lane]      = e0 * inv;
  out[(size_t)wid * 64 + lane + 32] = e1 * inv;
}

// ---------------- host orchestration ----------------
extern "C" void kernel_launch(void* const* d_in, const int* in_sizes, int n_in,
                              void* d_out, int out_size, void* d_ws, size_t ws_size,
                              hipStream_t stream) {
  (void)n_in; (void)out_size; (void)ws_size;
  const float* x       = (const float*)d_in[0];
  const int*   edges   = (const int*)d_in[1];
  const float* layer_W = (const float*)d_in[2];
  const float* layer_b = (const float*)d_in[3];
  const float* heads_W = (const float*)d_in[4];
  const float* heads_a = (const float*)d_in[5];
  const float* end_W   = (const float*)d_in[6];
  const float* end_a   = (const float*)d_in[7];
  float* out = (float*)d_out;

  const int Nn = in_sizes[0] / ND;   // 50000
  const int E  = in_sizes[1] / 2;    // 1600000
  const int* srcI = edges;
  const int* dstI = edges + E;

  float* ws = (float*)d_ws;
  size_t off = 0;
  float* Wpk    = ws + off; off += (size_t)ND * HC;
  float* h1     = ws + off; off += (size_t)Nn * ND;
  float* hh     = ws + off; off += (size_t)Nn * HC;
  float* hagg   = ws + off; off += (size_t)Nn * HC;
  float* ssrc   = ws + off; off += (size_t)NH * Nn;
  float* sdst   = ws + off; off += (size_t)NH * Nn;
  float* mArr   = ws + off; off += (size_t)NH * Nn;
  float* denArr = ws + off; off += (size_t)NH * Nn;
  int*   cnt    = (int*)(ws + off); off += (size_t)Nn;
  int*   offs   = (int*)(ws + off); off += (size_t)Nn + 1;
  int*   sSrc   = (int*)(ws + off); off += (size_t)E;
  float* h3     = h1;                          // reuse (h1 dead after GEMM2)
  float* h3agg  = h1 + (size_t)Nn * HK;

  const dim3 blk(256);
  const dim3 gE((E + 255) / 256);

  // ---- counting sort of edges by destination (reused by both GAT layers) ----
  (void)hipMemsetAsync(cnt, 0, (size_t)Nn * sizeof(int), stream);
  edge_count_kernel<<<gE, blk, 0, stream>>>(dstI, cnt, E);
  scan_kernel<<<dim3(1), dim3(1024), 0, stream>>>(cnt, offs, Nn);
  (void)hipMemsetAsync(cnt, 0, (size_t)Nn * sizeof(int), stream);
  edge_scatter_kernel<<<gE, blk, 0, stream>>>(srcI, dstI, offs, cnt, sSrc, E);

  // ---- dense front-end (WMMA fp32) ----
  pack_headsW<<<dim3((ND * HC + 255) / 256), blk, 0, stream>>>(heads_W, Wpk);
  gemm_f32_wmma<<<dim3((Nn + TM - 1) / TM, ND / TN), blk, 0, stream>>>(
      x, layer_W, layer_b, h1, Nn, ND, ND);
  gemm_f32_wmma<<<dim3((Nn + TM - 1) / TM, HC / TN), blk, 0, stream>>>(
      h1, Wpk, nullptr, hh, Nn, ND, HC);

  // ---- heads: segment softmax + atomic-free aggregation (+fused ELU) ----
  proj_kernel<<<dim3((Nn * NH + 255) / 256), blk, 0, stream>>>(hh, heads_a, ssrc, sdst, Nn, NH);
  seg_softmax_kernel<<<dim3((Nn * NH + 255) / 256), blk, 0, stream>>>(
      offs, sSrc, ssrc, sdst, mArr, denArr, Nn, NH);
  seg_agg_kernel<<<dim3(Nn), blk, 0, stream>>>(
      offs, sSrc, ssrc, sdst, mArr, denArr, hh, hagg, Nn, NH, /*do_elu=*/1);

  // ---- final layer ----
  gemm_f32_wmma<<<dim3((Nn + TM - 1) / TM, HK / TN), blk, 0, stream>>>(
      hagg, end_W, nullptr, h3, Nn, HC, HK);
  proj_kernel<<<dim3((Nn + 255) / 256), blk, 0, stream>>>(h3, end_a, ssrc, sdst, Nn, 1);
  seg_softmax_kernel<<<dim3((Nn + 255) / 256), blk, 0, stream>>>(
      offs, sSrc, ssrc, sdst, mArr, denArr, Nn, 1);
  seg_agg_kernel<<<dim3(Nn), blk, 0, stream>>>(
      offs, sSrc, ssrc, sdst, mArr, denArr, h3, h3agg, Nn, 1, /*do_elu=*/0);

  // ---- ELU + row softmax -> d_out ----
  elu_softmax_kernel<<<dim3(((size_t)Nn * 32 + 255) / 256), blk, 0, stream>>>(h3agg, out, Nn);
}